// decompress_jpeg_64828236366478
// MI455X (gfx1250) — compile-verified
//
#include <hip/hip_runtime.h>

// CDNA5 / gfx1250, wave32. JPEG decompress: dequant -> 8x8 IDCT (via
// V_WMMA_F32_16X16X4_F32, separable form) -> block merge -> chroma 2x2
// upsample -> YCbCr->RGB -> clamp/scale.
//
// One wave handles one 16x16 luma pixel tile (4 luma blocks) plus the one
// aligned Cb and Cr 8x8 block covering it. IDCT per 16x16 packed matrix:
//   stage1: T = P * diag(G,G)        (4x wmma f32 16x16x4, K-accumulate)
//   stage2: OUT^T = T^T * diag(G,G)  (T^T built with 8 shfl_xor half-swaps)
// G[x][u] = 0.5*a(x)*cos((2u+1)*x*pi/16) folds ALPHA and the 0.25 factor.

typedef __attribute__((ext_vector_type(2))) float v2f;
typedef __attribute__((ext_vector_type(8))) float v8f;

static __device__ __constant__ float YQ[64] = {
  16,11,10,16,24,40,51,61,
  12,12,14,19,26,58,60,55,
  14,13,16,24,40,57,69,56,
  14,17,22,29,51,87,80,62,
  18,22,37,56,68,109,103,77,
  24,35,55,64,81,104,113,92,
  49,64,78,87,103,121,120,101,
  72,92,95,98,112,100,103,99 };

static __device__ __constant__ float CQ[64] = {
  17,18,24,47,99,99,99,99,
  18,21,26,66,99,99,99,99,
  24,26,56,99,99,99,99,99,
  47,66,99,99,99,99,99,99,
  99,99,99,99,99,99,99,99,
  99,99,99,99,99,99,99,99,
  99,99,99,99,99,99,99,99,
  99,99,99,99,99,99,99,99 };

#define WMMA_F32X4(A, Bv, Cc) \
  __builtin_amdgcn_wmma_f32_16x16x4_f32(false, (A), false, (Bv), (short)0, (Cc), false, false)

__device__ __forceinline__ v8f idct_stage1(const v2f* A, const v2f* bc) {
  v8f T = {0.f,0.f,0.f,0.f,0.f,0.f,0.f,0.f};
  T = WMMA_F32X4(A[0], bc[0], T);
  T = WMMA_F32X4(A[1], bc[1], T);
  T = WMMA_F32X4(A[2], bc[2], T);
  T = WMMA_F32X4(A[3], bc[3], T);
  return T;
}

// Build A = T^T from T (in C/D layout) and run stage 2: D = T^T * diag(G,G).
__device__ __forceinline__ v8f idct_stage2(v8f T, const v2f* bc, bool lo) {
  float a[8], s[8];
#pragma unroll
  for (int i = 0; i < 8; ++i) { a[i] = T[i]; s[i] = __shfl_xor(a[i], 16, 32); }
  v2f A0 = { lo ? a[0] : s[2], lo ? a[1] : s[3] };   // K chunk 0..3
  v2f A1 = { lo ? a[4] : s[6], lo ? a[5] : s[7] };   // K chunk 4..7
  v2f A2 = { lo ? s[0] : a[2], lo ? s[1] : a[3] };   // K chunk 8..11
  v2f A3 = { lo ? s[4] : a[6], lo ? s[5] : a[7] };   // K chunk 12..15
  v8f D = {0.f,0.f,0.f,0.f,0.f,0.f,0.f,0.f};
  D = WMMA_F32X4(A0, bc[0], D);
  D = WMMA_F32X4(A1, bc[1], D);
  D = WMMA_F32X4(A2, bc[2], D);
  D = WMMA_F32X4(A3, bc[3], D);
  return D;
}

__global__ __launch_bounds__(256) void decompress_jpeg_kernel(
    const float* __restrict__ ycoef, const float* __restrict__ cbcoef,
    const float* __restrict__ crcoef, const float* __restrict__ facp,
    float* __restrict__ out)
{
  // Fixed problem shape: B=16, H=W=1024 -> 64x64 tiles of 16x16 px per image.
  const int lane = threadIdx.x & 31;
  const int tile = blockIdx.x * 8 + (threadIdx.x >> 5);
  const int b  = tile >> 12;          // batch
  const int ty = (tile >> 6) & 63;    // tile row
  const int tx = tile & 63;           // tile col

  const int  M    = lane & 15;        // A/B/D row index (and B's column index n)
  const int  half = lane >> 4;
  const bool lo   = (half == 0);
  const float f   = facp[0];

  // ---- constant operand Bc = diag(G,G) in WMMA B layout (K = 2*half + j per chunk)
  v2f bc[4];
#pragma unroll
  for (int t = 0; t < 4; ++t) {
#pragma unroll
    for (int j = 0; j < 2; ++j) {
      int k = 4*t + 2*half + j;
      float g = 0.0f;
      if ((k >> 3) == (M >> 3)) {
        int x = k & 7, u = M & 7;
        float ax = (x == 0) ? 0.70710678118654752f : 1.0f;
        g = 0.5f * ax * __cosf((float)((2*u + 1) * x) * 0.19634954084936207f);
      }
      bc[t][j] = g;
    }
  }

  // ---- load + dequantize luma A operand: P = [[S00,S01],[S10,S11]]
  const int xr = M & 7;                          // row within 8x8 block
  const size_t ybase =
      (((size_t)b * 16384) + (size_t)(2*ty + (M >> 3)) * 128 + 2*tx) * 64 + (size_t)xr * 8;
  const float* pL = ycoef + ybase;               // left block row; +64 = right block
  float2 l0 = *(const float2*)(pL +       2*half);
  float2 l1 = *(const float2*)(pL +  4 +  2*half);
  float2 l2 = *(const float2*)(pL + 64 +  2*half);
  float2 l3 = *(const float2*)(pL + 68 +  2*half);
  const float q0 = YQ[xr*8 + 2*half    ] * f;
  const float q1 = YQ[xr*8 + 2*half + 1] * f;
  const float q2 = YQ[xr*8 + 2*half + 4] * f;
  const float q3 = YQ[xr*8 + 2*half + 5] * f;
  v2f AL[4] = { { l0.x*q0, l0.y*q1 }, { l1.x*q2, l1.y*q3 },
                { l2.x*q0, l2.y*q1 }, { l3.x*q2, l3.y*q3 } };

  // ---- load + dequantize chroma A operand: Pc = [[Cb,Cr],[0,0]]
  const size_t cbase =
      (((size_t)b * 4096) + (size_t)ty * 64 + tx) * 64 + (size_t)xr * 8;
  const float* pB = cbcoef + cbase;
  const float* pR = crcoef + cbase;
  float2 c0 = *(const float2*)(pB +      2*half);
  float2 c1 = *(const float2*)(pB + 4 +  2*half);
  float2 c2 = *(const float2*)(pR +      2*half);
  float2 c3 = *(const float2*)(pR + 4 +  2*half);
  const float fm = (M < 8) ? f : 0.0f;           // zero rows 8..15
  const float p0 = CQ[xr*8 + 2*half    ] * fm;
  const float p1 = CQ[xr*8 + 2*half + 1] * fm;
  const float p2 = CQ[xr*8 + 2*half + 4] * fm;
  const float p3 = CQ[xr*8 + 2*half + 5] * fm;
  v2f AC[4] = { { c0.x*p0, c0.y*p1 }, { c1.x*p2, c1.y*p3 },
                { c2.x*p0, c2.y*p1 }, { c3.x*p2, c3.y*p3 } };

  // ---- two-stage IDCT via WMMA (16 wmma ops total per wave)
  v8f DL = idct_stage2(idct_stage1(AL, bc), bc, lo);   // OUT^T luma: lane=(16*half? , ) holds pixel(row=M, col=8*half+j) in DL[j]
  v8f DC = idct_stage2(idct_stage1(AC, bc), bc, lo);   // DC[j]: lane u (<16) = cb[u][j], lane 16+u = cr[u][j]

  // ---- chroma gather for 2x2 nearest upsample (cb/cr col = (8*half+j)>>1, row = M>>1)
  const int us = M >> 1;
  float cb_lo[4], cb_hi[4], cr_lo[4], cr_hi[4];
#pragma unroll
  for (int m = 0; m < 4; ++m) {
    float dlo = DC[m], dhi = DC[4 + m];
    cb_lo[m] = __shfl(dlo, us,      32);
    cb_hi[m] = __shfl(dhi, us,      32);
    cr_lo[m] = __shfl(dlo, 16 + us, 32);
    cr_hi[m] = __shfl(dhi, 16 + us, 32);
  }

  // ---- color convert, clamp, scale. Chroma +128 (IDCT) cancels -128 (shift).
  float R[8], G8[8], B8[8];
#pragma unroll
  for (int j = 0; j < 8; ++j) {
    int m = j >> 1;
    float cbv = lo ? cb_lo[m] : cb_hi[m];
    float crv = lo ? cr_lo[m] : cr_hi[m];
    float Y = DL[j] + 128.0f;
    float r  = Y + 1.402f   * crv;
    float g  = Y - 0.344136f * cbv - 0.714136f * crv;
    float bb = Y + 1.772f   * cbv;
    R[j]  = fminf(fmaxf(r,  0.0f), 255.0f) * (1.0f/255.0f);
    G8[j] = fminf(fmaxf(g,  0.0f), 255.0f) * (1.0f/255.0f);
    B8[j] = fminf(fmaxf(bb, 0.0f), 255.0f) * (1.0f/255.0f);
  }

  // ---- store: each lane writes 8 contiguous pixels of one row per plane
  const int row = ty * 16 + M;
  const int col = tx * 16 + 8 * half;
  size_t o = (((size_t)b * 3) * 1024 + (size_t)row) * 1024 + (size_t)col;
  float4* pr = (float4*)(out + o);
  pr[0] = make_float4(R[0], R[1], R[2], R[3]);
  pr[1] = make_float4(R[4], R[5], R[6], R[7]);
  float4* pg = (float4*)(out + o + 1048576);
  pg[0] = make_float4(G8[0], G8[1], G8[2], G8[3]);
  pg[1] = make_float4(G8[4], G8[5], G8[6], G8[7]);
  float4* pb = (float4*)(out + o + 2097152);
  pb[0] = make_float4(B8[0], B8[1], B8[2], B8[3]);
  pb[1] = make_float4(B8[4], B8[5], B8[6], B8[7]);
}

extern "C" void kernel_launch(void* const* d_in, const int* in_sizes, int n_in,
                              void* d_out, int out_size, void* d_ws, size_t ws_size,
                              hipStream_t stream) {
  const float* y  = (const float*)d_in[0];
  const float* cb = (const float*)d_in[1];
  const float* cr = (const float*)d_in[2];
  const float* fc = (const float*)d_in[3];   // scalar factor (device)
  float* out = (float*)d_out;
  // 16 batches * 64*64 tiles = 65536 waves; 8 waves (256 threads) per block.
  dim3 grid(65536 / 8), block(256);
  hipLaunchKernelGGL(decompress_jpeg_kernel, grid, block, 0, stream, y, cb, cr, fc, out);
}